// SignedGraphConvolutionalNetwork2_91190745628895
// MI455X (gfx1250) — compile-verified
//
#include <hip/hip_runtime.h>
#include <hip/hip_bf16.h>
#include <math.h>

typedef __attribute__((ext_vector_type(2))) float v2f;
typedef __attribute__((ext_vector_type(8))) float v8f;

#define D_IN 300
#define HDIM 64

// ---------------------------------------------------------------------------
// Utility kernels
// ---------------------------------------------------------------------------
__global__ void fill_zero_kernel(float* __restrict__ p, long long n) {
    long long i = (long long)blockIdx.x * blockDim.x + threadIdx.x;
    if (i < n) p[i] = 0.0f;
}

__global__ void degree_kernel(const int* __restrict__ edges, float* __restrict__ deg, int E) {
    int e = blockIdx.x * blockDim.x + threadIdx.x;
    if (e < E) atomicAdd(&deg[edges[e]], 1.0f);   // row = edges[0][e]
}

__global__ void invdeg_kernel(float* __restrict__ deg, int n) {
    int i = blockIdx.x * blockDim.x + threadIdx.x;
    if (i < n) deg[i] = 1.0f / fmaxf(deg[i], 1.0f);
}

// scatter-sum: for each edge i: out[row[i]] += x[col[i]]  (row=edges[0..E), col=edges[E..2E))
__global__ void scatter_sum_kernel(const float* __restrict__ x, const int* __restrict__ edges,
                                   float* __restrict__ out, int E, int D) {
    long long idx = (long long)blockIdx.x * blockDim.x + threadIdx.x;
    long long total = (long long)E * D;
    if (idx >= total) return;
    int e = (int)(idx / D);
    int f = (int)(idx % D);
    int r = edges[e];
    int c = edges[E + e];
    atomicAdd(&out[(long long)r * D + f], x[(long long)c * D + f]);
}

// ---------------------------------------------------------------------------
// WMMA helpers (V_WMMA_F32_16X16X4_F32, wave32)
// A 16x4 f32: lane<16 -> A[M=lane][K=0..1], lane>=16 -> A[M=lane-16][K=2..3]
// B 4x16 f32: lane<16 -> B[K=0..1][N=lane], lane>=16 -> B[K=2..3][N=lane-16]
// C/D 16x16 f32: vgpr v, lane L: C[v + 8*(L/16)][L%16]
// ---------------------------------------------------------------------------
__device__ __forceinline__ v8f wmma4(v2f a, v2f b, v8f c) {
    return __builtin_amdgcn_wmma_f32_16x16x4_f32(false, a, false, b, (short)0, c, false, false);
}

// One k-step (4 deep) against weight matrix W[600|192][64]; Wk = &W[kk*64]
__device__ __forceinline__ void wmma_step_w(v8f acc[4], v2f a, const float* __restrict__ Wk, int lr) {
#pragma unroll
    for (int j = 0; j < 4; ++j) {
        v2f b;
        b.x = Wk[j * 16 + lr];
        b.y = Wk[HDIM + j * 16 + lr];
        acc[j] = wmma4(a, b, acc[j]);
    }
}

// bias add + per-row l2norm + tanh + store of a 16x64 strip (rows m0..m0+15)
__device__ __forceinline__ void strip_epilogue(v8f acc[4], const float* __restrict__ bias,
                                               float* __restrict__ out, int m0, int lr, int half) {
    float bj[4];
#pragma unroll
    for (int j = 0; j < 4; ++j) bj[j] = bias[j * 16 + lr];
#pragma unroll
    for (int v = 0; v < 8; ++v) {
        float s = 0.0f;
#pragma unroll
        for (int j = 0; j < 4; ++j) {
            float t = acc[j][v] + bj[j];
            acc[j][v] = t;
            s += t * t;
        }
        // rows 0..7 live on lanes 0..15, rows 8..15 on lanes 16..31: reduce within half
        s += __shfl_xor(s, 1);
        s += __shfl_xor(s, 2);
        s += __shfl_xor(s, 4);
        s += __shfl_xor(s, 8);
        float inv = 1.0f / fmaxf(sqrtf(s), 1e-12f);
        int row = m0 + v + 8 * half;
#pragma unroll
        for (int j = 0; j < 4; ++j)
            out[(long long)row * HDIM + j * 16 + lr] = tanhf(acc[j][v] * inv);
    }
}

// ---------------------------------------------------------------------------
// Base SAGE layer: h = tanh(l2norm([agg/deg, X] @ W + b)), K = 600
// one wave computes a 16-row x 64-col strip
// ---------------------------------------------------------------------------
__global__ void gemm_base_kernel(const float* __restrict__ aggsum, const float* __restrict__ invdeg,
                                 const float* __restrict__ X, const float* __restrict__ W,
                                 const float* __restrict__ bias, float* __restrict__ out, int N) {
    int wave = (blockIdx.x * blockDim.x + threadIdx.x) >> 5;
    int lane = threadIdx.x & 31;
    int half = lane >> 4;
    int lr   = lane & 15;
    int koff = half * 2;
    int m0 = wave * 16;
    if (m0 >= N) return;
    int row = m0 + lr;
    float s0 = invdeg[row];
    const float* arow = aggsum + (long long)row * D_IN;
    const float* xrow = X + (long long)row * D_IN;

    const v8f vzero = {0.f, 0.f, 0.f, 0.f, 0.f, 0.f, 0.f, 0.f};
    v8f acc[4] = {vzero, vzero, vzero, vzero};

    // segment 0: agg (scaled by 1/deg), k in [0,300)
    for (int k0 = 0; k0 < D_IN; k0 += 4) {
        v2f a = *(const v2f*)(arow + k0 + koff);
        a.x *= s0; a.y *= s0;
        wmma_step_w(acc, a, W + (long long)(k0 + koff) * HDIM, lr);
    }
    // segment 1: X, k in [300,600)
    for (int k0 = 0; k0 < D_IN; k0 += 4) {
        v2f a = *(const v2f*)(xrow + k0 + koff);
        wmma_step_w(acc, a, W + (long long)(D_IN + k0 + koff) * HDIM, lr);
    }
    strip_epilogue(acc, bias, out, m0, lr, half);
}

// ---------------------------------------------------------------------------
// Deep SAGE layer: h = tanh(l2norm([a1/deg1, a2/deg2, x1] @ W + b)), K = 192
// ---------------------------------------------------------------------------
__global__ void gemm_deep_kernel(const float* __restrict__ a1, const float* __restrict__ sc1,
                                 const float* __restrict__ a2, const float* __restrict__ sc2,
                                 const float* __restrict__ x1, const float* __restrict__ W,
                                 const float* __restrict__ bias, float* __restrict__ out, int N) {
    int wave = (blockIdx.x * blockDim.x + threadIdx.x) >> 5;
    int lane = threadIdx.x & 31;
    int half = lane >> 4;
    int lr   = lane & 15;
    int koff = half * 2;
    int m0 = wave * 16;
    if (m0 >= N) return;
    int row = m0 + lr;
    float s1 = sc1[row], s2 = sc2[row];
    const float* p1 = a1 + (long long)row * HDIM;
    const float* p2 = a2 + (long long)row * HDIM;
    const float* p3 = x1 + (long long)row * HDIM;

    const v8f vzero = {0.f, 0.f, 0.f, 0.f, 0.f, 0.f, 0.f, 0.f};
    v8f acc[4] = {vzero, vzero, vzero, vzero};

#pragma unroll
    for (int k0 = 0; k0 < HDIM; k0 += 4) {
        v2f a = *(const v2f*)(p1 + k0 + koff);
        a.x *= s1; a.y *= s1;
        wmma_step_w(acc, a, W + (long long)(k0 + koff) * HDIM, lr);
    }
#pragma unroll
    for (int k0 = 0; k0 < HDIM; k0 += 4) {
        v2f a = *(const v2f*)(p2 + k0 + koff);
        a.x *= s2; a.y *= s2;
        wmma_step_w(acc, a, W + (long long)(HDIM + k0 + koff) * HDIM, lr);
    }
#pragma unroll
    for (int k0 = 0; k0 < HDIM; k0 += 4) {
        v2f a = *(const v2f*)(p3 + k0 + koff);
        wmma_step_w(acc, a, W + (long long)(2 * HDIM + k0 + koff) * HDIM, lr);
    }
    strip_epilogue(acc, bias, out, m0, lr, half);
}

// ---------------------------------------------------------------------------
// X_mol = l2norm(concat(h_pos2, h_neg2)); one wave per row (128 cols)
// ---------------------------------------------------------------------------
__global__ void xmol_kernel(const float* __restrict__ hp, const float* __restrict__ hn,
                            float* __restrict__ Xmol, int N) {
    int wave = (blockIdx.x * blockDim.x + threadIdx.x) >> 5;
    int lane = threadIdx.x & 31;
    if (wave >= N) return;
    v2f p = *(const v2f*)(hp + (long long)wave * HDIM + 2 * lane);
    v2f q = *(const v2f*)(hn + (long long)wave * HDIM + 2 * lane);
    float s = p.x * p.x + p.y * p.y + q.x * q.x + q.y * q.y;
    s += __shfl_xor(s, 1);
    s += __shfl_xor(s, 2);
    s += __shfl_xor(s, 4);
    s += __shfl_xor(s, 8);
    s += __shfl_xor(s, 16);
    float inv = 1.0f / fmaxf(sqrtf(s), 1e-12f);
    float* o = Xmol + (long long)wave * 128;
    v2f op; op.x = p.x * inv; op.y = p.y * inv;
    v2f oq; oq.x = q.x * inv; oq.y = q.y * inv;
    *(v2f*)(o + 2 * lane) = op;
    *(v2f*)(o + HDIM + 2 * lane) = oq;
}

// ---------------------------------------------------------------------------
// pred = (X_mol @ X_mol^T) * mask ; loss partial sums.
// One wave -> 16x64 output tile (4 accumulators, A fragment reused 4x), K = 128.
// grid = (N/256, N/16), block = 128 (4 waves).
// ---------------------------------------------------------------------------
__global__ void pred_loss_kernel(const float* __restrict__ Xmol, const float* __restrict__ mask,
                                 const float* __restrict__ labels, float* __restrict__ pred,
                                 float* __restrict__ lacc, int N) {
    int wave = threadIdx.x >> 5;
    int lane = threadIdx.x & 31;
    int half = lane >> 4;
    int lr   = lane & 15;
    int koff = half * 2;
    int m0 = blockIdx.y * 16;
    int n0 = (blockIdx.x * 4 + wave) * 64;

    const float* Arow = Xmol + (long long)(m0 + lr) * 128;
    const float* Brow[4];
#pragma unroll
    for (int j = 0; j < 4; ++j) Brow[j] = Xmol + (long long)(n0 + j * 16 + lr) * 128;

    const v8f vzero = {0.f, 0.f, 0.f, 0.f, 0.f, 0.f, 0.f, 0.f};
    v8f acc[4] = {vzero, vzero, vzero, vzero};

#pragma unroll 4
    for (int k0 = 0; k0 < 128; k0 += 4) {
        v2f a = *(const v2f*)(Arow + k0 + koff);
#pragma unroll
        for (int j = 0; j < 4; ++j) {
            v2f b = *(const v2f*)(Brow[j] + k0 + koff);
            acc[j] = wmma4(a, b, acc[j]);
        }
    }

    float local = 0.0f;
#pragma unroll
    for (int v = 0; v < 8; ++v) {
        int row = m0 + v + 8 * half;
        long long base = (long long)row * N;
#pragma unroll
        for (int j = 0; j < 4; ++j) {
            long long idx = base + n0 + j * 16 + lr;
            float p = acc[j][v] * mask[idx];
            pred[idx] = p;
            float d = p - labels[idx];
            local += d * d;
        }
    }
    local += __shfl_xor(local, 1);
    local += __shfl_xor(local, 2);
    local += __shfl_xor(local, 4);
    local += __shfl_xor(local, 8);
    local += __shfl_xor(local, 16);
    if (lane == 0) atomicAdd(lacc, local);
}

__global__ void finalize_kernel(const float* __restrict__ lacc, float* __restrict__ out, int N) {
    if (threadIdx.x == 0 && blockIdx.x == 0)
        out[0] = lacc[0] / ((float)N * (float)N);
}

// ---------------------------------------------------------------------------
// Launcher
// ---------------------------------------------------------------------------
extern "C" void kernel_launch(void* const* d_in, const int* in_sizes, int n_in,
                              void* d_out, int out_size, void* d_ws, size_t ws_size,
                              hipStream_t stream) {
    const float* X      = (const float*)d_in[0];
    const int*   pe     = (const int*)d_in[1];
    const int*   ne     = (const int*)d_in[2];
    const float* labels = (const float*)d_in[3];
    const float* mask   = (const float*)d_in[4];
    const float* Wpb = (const float*)d_in[5];
    const float* bpb = (const float*)d_in[6];
    const float* Wnb = (const float*)d_in[7];
    const float* bnb = (const float*)d_in[8];
    const float* Wpd = (const float*)d_in[9];
    const float* bpd = (const float*)d_in[10];
    const float* Wnd = (const float*)d_in[11];
    const float* bnd = (const float*)d_in[12];

    const int N = in_sizes[0] / D_IN;   // 8192
    const int E = in_sizes[1] / 2;      // 262144

    float* out  = (float*)d_out;
    float* Xmol = out + 1;
    float* pred = out + 1 + (long long)N * 128;

    // workspace layout (floats)
    float* ws      = (float*)d_ws;
    float* degpos  = ws;                       // N
    float* degneg  = degpos + N;               // N
    float* agg     = degneg + N;               // N*300 (reused pos then neg)
    float* hpos    = agg + (long long)N * D_IN;
    float* hneg    = hpos + (long long)N * HDIM;
    float* app     = hneg + (long long)N * HDIM;   // scatter(h_pos, pos)
    float* ann     = app + (long long)N * HDIM;    // scatter(h_neg, neg)
    float* anp     = ann + (long long)N * HDIM;    // scatter(h_neg, pos)
    float* apn     = anp + (long long)N * HDIM;    // scatter(h_pos, neg)
    float* hpos2   = apn + (long long)N * HDIM;
    float* hneg2   = hpos2 + (long long)N * HDIM;
    float* lacc    = hneg2 + (long long)N * HDIM;  // 1

    const int T = 256;
    auto blocks = [](long long n, int t) { return (unsigned)((n + t - 1) / t); };

    // zero degrees, agg, deep aggregates, loss accumulator
    fill_zero_kernel<<<blocks(2LL * N, T), T, 0, stream>>>(degpos, 2LL * N);
    fill_zero_kernel<<<blocks((long long)N * D_IN, T), T, 0, stream>>>(agg, (long long)N * D_IN);
    fill_zero_kernel<<<blocks(4LL * N * HDIM, T), T, 0, stream>>>(app, 4LL * N * HDIM);
    fill_zero_kernel<<<1, 32, 0, stream>>>(lacc, 1);

    // degrees -> 1/max(deg,1)
    degree_kernel<<<blocks(E, T), T, 0, stream>>>(pe, degpos, E);
    degree_kernel<<<blocks(E, T), T, 0, stream>>>(ne, degneg, E);
    invdeg_kernel<<<blocks(2LL * N, T), T, 0, stream>>>(degpos, 2 * N);

    const int strips = N / 16;                 // waves needed per GEMM
    const unsigned gemmBlocks = (unsigned)((strips * 32 + 127) / 128);

    // base pos
    scatter_sum_kernel<<<blocks((long long)E * D_IN, T), T, 0, stream>>>(X, pe, agg, E, D_IN);
    gemm_base_kernel<<<gemmBlocks, 128, 0, stream>>>(agg, degpos, X, Wpb, bpb, hpos, N);

    // base neg (reuse agg buffer)
    fill_zero_kernel<<<blocks((long long)N * D_IN, T), T, 0, stream>>>(agg, (long long)N * D_IN);
    scatter_sum_kernel<<<blocks((long long)E * D_IN, T), T, 0, stream>>>(X, ne, agg, E, D_IN);
    gemm_base_kernel<<<gemmBlocks, 128, 0, stream>>>(agg, degneg, X, Wnb, bnb, hneg, N);

    // deep aggregates
    scatter_sum_kernel<<<blocks((long long)E * HDIM, T), T, 0, stream>>>(hpos, pe, app, E, HDIM);
    scatter_sum_kernel<<<blocks((long long)E * HDIM, T), T, 0, stream>>>(hneg, ne, ann, E, HDIM);
    scatter_sum_kernel<<<blocks((long long)E * HDIM, T), T, 0, stream>>>(hneg, pe, anp, E, HDIM);
    scatter_sum_kernel<<<blocks((long long)E * HDIM, T), T, 0, stream>>>(hpos, ne, apn, E, HDIM);

    // deep layers
    gemm_deep_kernel<<<gemmBlocks, 128, 0, stream>>>(app, degpos, ann, degneg, hpos, Wpd, bpd, hpos2, N);
    gemm_deep_kernel<<<gemmBlocks, 128, 0, stream>>>(anp, degpos, apn, degneg, hneg, Wnd, bnd, hneg2, N);

    // X_mol
    xmol_kernel<<<(unsigned)((N * 32 + 255) / 256), 256, 0, stream>>>(hpos2, hneg2, Xmol, N);

    // pred + loss
    dim3 pgrid(N / 256, N / 16);
    pred_loss_kernel<<<pgrid, 128, 0, stream>>>(Xmol, mask, labels, pred, lacc, N);
    finalize_kernel<<<1, 32, 0, stream>>>(lacc, out, N);
}